// NodeFeatureEncoder_76347338654281
// MI455X (gfx1250) — compile-verified
//
#include <hip/hip_runtime.h>

#define NODES  40000
#define EDGES  640000
#define ETOT   (EDGES + NODES)   // edges + self loops
#define DIM    128
#define HEADS  8
#define CHN    16
#define LAYERS 3
#define NGRAPH 64
#define SLOPE  0.2f
#define EPS_LN 1e-5f

typedef __attribute__((ext_vector_type(2))) float v2f;
typedef __attribute__((ext_vector_type(8))) float v8f;

// ---------------- GEMM: H = X @ W (+bias). X:[M,128], W:[128,128] row-major ----------------
// One wave computes a 16x16 tile via 32x V_WMMA_F32_16X16X4_F32 (K stepped by 4).
__global__ __launch_bounds__(256)
void gemm128_wmma(const float* __restrict__ X, const float* __restrict__ W,
                  const float* __restrict__ bias, float* __restrict__ H)
{
    const int wave = threadIdx.x >> 5;
    const int lane = threadIdx.x & 31;
    const int half = lane >> 4;      // 0: K pair {0,1}; 1: K pair {2,3}
    const int l16  = lane & 15;
    const int m0   = blockIdx.x * 16;
    const int n0   = wave * 16;

    v8f acc = {};
    const float* Arow = X + (size_t)(m0 + l16) * DIM;

    #pragma unroll
    for (int k = 0; k < DIM; k += 4) {
        const int ka = k + 2 * half;
        // A 16x4: lanes 0-15 hold K={k,k+1} of row l16; lanes 16-31 hold K={k+2,k+3}
        float2 av = *(const float2*)(Arow + ka);
        v2f a; a.x = av.x; a.y = av.y;
        // B 4x16: mirrored layout, row-striped across lanes
        v2f b;
        b.x = W[(size_t)ka       * DIM + n0 + l16];
        b.y = W[(size_t)(ka + 1) * DIM + n0 + l16];
        acc = __builtin_amdgcn_wmma_f32_16x16x4_f32(false, a, false, b,
                                                    (short)0, acc, false, false);
    }

    const float bv = bias ? bias[n0 + l16] : 0.0f;
    // C/D layout: VGPR r -> row (r + 8*half), col l16
    #pragma unroll
    for (int r = 0; r < 8; ++r)
        H[(size_t)(m0 + r + 8 * half) * DIM + n0 + l16] = acc[r] + bv;
}

// ---------------- attention coefficients: al_src/al_dst = sum_c h[n,h,c] * a[h,c] ----------------
__global__ __launch_bounds__(256)
void attn_coef(const float* __restrict__ H, const float* __restrict__ a_s,
               const float* __restrict__ a_d, float* __restrict__ ALs,
               float* __restrict__ ALd)
{
    int t = blockIdx.x * blockDim.x + threadIdx.x;
    if (t >= NODES * HEADS) return;
    const int node = t >> 3, h = t & 7;
    const float* hp = H + (size_t)node * DIM + h * CHN;
    float s = 0.f, d = 0.f;
    #pragma unroll
    for (int c = 0; c < CHN; ++c) {
        s += hp[c] * a_s[h * CHN + c];
        d += hp[c] * a_d[h * CHN + c];
    }
    ALs[t] = s; ALd[t] = d;
}

// ---------------- per-layer scratch init ----------------
__global__ __launch_bounds__(256)
void layer_init(unsigned* __restrict__ MU, float* __restrict__ DEN,
                float* __restrict__ ACC)
{
    int i = blockIdx.x * blockDim.x + threadIdx.x;
    if (i < NODES * DIM) ACC[i] = 0.f;
    if (i < NODES * HEADS) { MU[i] = 0x007FFFFFu; /* key(-inf) */ DEN[i] = 0.f; }
}

__device__ __forceinline__ float lrelu(float z) { return z > 0.f ? z : SLOPE * z; }
__device__ __forceinline__ int esrc(const int* es, int e) { return (e < EDGES) ? es[e] : (e - EDGES); }
__device__ __forceinline__ int edst(const int* ed, int e) { return (e < EDGES) ? ed[e] : (e - EDGES); }

// ---------------- pass 1: segment max (order-preserving uint atomicMax) ----------------
__global__ __launch_bounds__(256)
void edge_max_kernel(const int* __restrict__ es, const int* __restrict__ ed,
                     const float* __restrict__ ALs, const float* __restrict__ ALd,
                     unsigned* __restrict__ MU)
{
    int e = blockIdx.x * blockDim.x + threadIdx.x;
    if (e >= ETOT) return;
    const int s = esrc(es, e), d = edst(ed, e);
    #pragma unroll
    for (int h = 0; h < HEADS; ++h) {
        float z = lrelu(ALs[s * HEADS + h] + ALd[d * HEADS + h]);
        unsigned u = __float_as_uint(z);
        unsigned key = (u & 0x80000000u) ? ~u : (u | 0x80000000u);
        atomicMax(&MU[d * HEADS + h], key);
    }
}

__global__ __launch_bounds__(256)
void decode_max(unsigned* __restrict__ MU)
{
    int i = blockIdx.x * blockDim.x + threadIdx.x;
    if (i >= NODES * HEADS) return;
    unsigned k = MU[i];
    unsigned u = (k & 0x80000000u) ? (k & 0x7FFFFFFFu) : ~k;
    MU[i] = u;   // now holds float bits of the max
}

// ---------------- pass 2: denom = segment_sum(exp(logit - m[dst])) ----------------
__global__ __launch_bounds__(256)
void edge_sum_kernel(const int* __restrict__ es, const int* __restrict__ ed,
                     const float* __restrict__ ALs, const float* __restrict__ ALd,
                     const float* __restrict__ M, float* __restrict__ DEN)
{
    int e = blockIdx.x * blockDim.x + threadIdx.x;
    if (e >= ETOT) return;
    const int s = esrc(es, e), d = edst(ed, e);
    #pragma unroll
    for (int h = 0; h < HEADS; ++h) {
        float z = lrelu(ALs[s * HEADS + h] + ALd[d * HEADS + h]);
        atomicAdd(&DEN[d * HEADS + h], expf(z - M[d * HEADS + h]));
    }
}

// ---------------- pass 3: weighted message aggregation ----------------
__global__ __launch_bounds__(256)
void edge_msg_kernel(const int* __restrict__ es, const int* __restrict__ ed,
                     const float* __restrict__ ALs, const float* __restrict__ ALd,
                     const float* __restrict__ M, const float* __restrict__ DEN,
                     const float* __restrict__ H, float* __restrict__ ACC)
{
    long long t = (long long)blockIdx.x * blockDim.x + threadIdx.x;
    if (t >= (long long)ETOT * HEADS) return;
    const int e = (int)(t >> 3), h = (int)(t & 7);
    const int s = esrc(es, e), d = edst(ed, e);
    float z = lrelu(ALs[s * HEADS + h] + ALd[d * HEADS + h]);
    float ex = expf(z - M[d * HEADS + h]);
    float alpha = ex / (DEN[d * HEADS + h] + 1e-16f);

    const float4* hp = (const float4*)(H + (size_t)s * DIM + h * CHN);
    float* op = ACC + (size_t)d * DIM + h * CHN;
    #pragma unroll
    for (int q = 0; q < 4; ++q) {
        float4 v = hp[q];
        atomicAdd(&op[q * 4 + 0], alpha * v.x);
        atomicAdd(&op[q * 4 + 1], alpha * v.y);
        atomicAdd(&op[q * 4 + 2], alpha * v.z);
        atomicAdd(&op[q * 4 + 3], alpha * v.w);
    }
}

// ---------------- residual + bias + LayerNorm + ReLU (one wave32 per node) ----------------
__global__ __launch_bounds__(256)
void post_kernel(float* __restrict__ X, const float* __restrict__ ACC,
                 const float* __restrict__ bias, const float* __restrict__ gamma,
                 const float* __restrict__ beta)
{
    const int node = blockIdx.x * 8 + (threadIdx.x >> 5);
    const int lane = threadIdx.x & 31;
    float4 a = *(const float4*)(ACC + (size_t)node * DIM + lane * 4);
    float4 x = *(const float4*)(X   + (size_t)node * DIM + lane * 4);
    float4 b = *(const float4*)(bias + lane * 4);
    float4 v;
    v.x = x.x + a.x + b.x; v.y = x.y + a.y + b.y;
    v.z = x.z + a.z + b.z; v.w = x.w + a.w + b.w;

    float sum = v.x + v.y + v.z + v.w;
    float sq  = v.x * v.x + v.y * v.y + v.z * v.z + v.w * v.w;
    #pragma unroll
    for (int off = 16; off >= 1; off >>= 1) {
        sum += __shfl_xor(sum, off, 32);
        sq  += __shfl_xor(sq,  off, 32);
    }
    const float mu  = sum * (1.0f / DIM);
    const float var = sq * (1.0f / DIM) - mu * mu;
    const float inv = rsqrtf(var + EPS_LN);

    float4 g = *(const float4*)(gamma + lane * 4);
    float4 be = *(const float4*)(beta + lane * 4);
    float4 y;
    y.x = fmaxf(0.f, (v.x - mu) * inv * g.x + be.x);
    y.y = fmaxf(0.f, (v.y - mu) * inv * g.y + be.y);
    y.z = fmaxf(0.f, (v.z - mu) * inv * g.z + be.z);
    y.w = fmaxf(0.f, (v.w - mu) * inv * g.w + be.w);
    *(float4*)(X + (size_t)node * DIM + lane * 4) = y;
}

// ---------------- global mean pool ----------------
__global__ __launch_bounds__(256)
void pool_init(float* __restrict__ GS, float* __restrict__ CNT)
{
    int i = blockIdx.x * blockDim.x + threadIdx.x;
    if (i < NGRAPH * DIM) GS[i] = 0.f;
    if (i < NGRAPH) CNT[i] = 0.f;
}

__global__ __launch_bounds__(256)
void pool_accum(const float* __restrict__ X, const int* __restrict__ batch,
                float* __restrict__ GS, float* __restrict__ CNT)
{
    int t = blockIdx.x * blockDim.x + threadIdx.x;
    if (t >= NODES * 32) return;
    const int node = t >> 5, lane = t & 31;
    const int g = batch[node];
    float4 v = *(const float4*)(X + (size_t)node * DIM + lane * 4);
    atomicAdd(&GS[g * DIM + lane * 4 + 0], v.x);
    atomicAdd(&GS[g * DIM + lane * 4 + 1], v.y);
    atomicAdd(&GS[g * DIM + lane * 4 + 2], v.z);
    atomicAdd(&GS[g * DIM + lane * 4 + 3], v.w);
    if (lane == 0) atomicAdd(&CNT[g], 1.0f);
}

__global__ __launch_bounds__(256)
void pool_final(const float* __restrict__ GS, const float* __restrict__ CNT,
                float* __restrict__ out)
{
    int i = blockIdx.x * blockDim.x + threadIdx.x;
    if (i >= NGRAPH * DIM) return;
    out[i] = GS[i] / fmaxf(CNT[i >> 7], 1.0f);
}

extern "C" void kernel_launch(void* const* d_in, const int* in_sizes, int n_in,
                              void* d_out, int out_size, void* d_ws, size_t ws_size,
                              hipStream_t stream)
{
    const float* x_in   = (const float*)d_in[0];
    const int*   ei     = (const int*)  d_in[1];
    const int*   batch  = (const int*)  d_in[2];
    const float* W_in   = (const float*)d_in[3];
    const float* b_in   = (const float*)d_in[4];
    const float* W_l    = (const float*)d_in[5];
    const float* att_s  = (const float*)d_in[6];
    const float* att_d  = (const float*)d_in[7];
    const float* bias_l = (const float*)d_in[8];
    const float* gamma  = (const float*)d_in[9];
    const float* beta   = (const float*)d_in[10];
    const int* es = ei;
    const int* ed = ei + EDGES;

    float* WX  = (float*)d_ws;                 // current node features [N,128]
    float* WH  = WX  + (size_t)NODES * DIM;    // h = x @ W             [N,128]
    float* WA  = WH  + (size_t)NODES * DIM;    // aggregation acc       [N,128]
    float* ALs = WA  + (size_t)NODES * DIM;    // [N,8]
    float* ALd = ALs + (size_t)NODES * HEADS;  // [N,8]
    float* M   = ALd + (size_t)NODES * HEADS;  // [N,8] (uint keys -> float max)
    float* DEN = M   + (size_t)NODES * HEADS;  // [N,8]
    float* GS  = DEN + (size_t)NODES * HEADS;  // [64,128]
    float* CNT = GS  + (size_t)NGRAPH * DIM;   // [64]

    // input projection: x = x @ W_in + b_in
    gemm128_wmma<<<NODES / 16, 256, 0, stream>>>(x_in, W_in, b_in, WX);

    for (int l = 0; l < LAYERS; ++l) {
        gemm128_wmma<<<NODES / 16, 256, 0, stream>>>(WX, W_l + (size_t)l * DIM * DIM, nullptr, WH);
        attn_coef<<<(NODES * HEADS + 255) / 256, 256, 0, stream>>>(
            WH, att_s + l * HEADS * CHN, att_d + l * HEADS * CHN, ALs, ALd);
        layer_init<<<(NODES * DIM + 255) / 256, 256, 0, stream>>>((unsigned*)M, DEN, WA);
        edge_max_kernel<<<(ETOT + 255) / 256, 256, 0, stream>>>(es, ed, ALs, ALd, (unsigned*)M);
        decode_max<<<(NODES * HEADS + 255) / 256, 256, 0, stream>>>((unsigned*)M);
        edge_sum_kernel<<<(ETOT + 255) / 256, 256, 0, stream>>>(es, ed, ALs, ALd, M, DEN);
        edge_msg_kernel<<<(ETOT * HEADS + 255) / 256, 256, 0, stream>>>(es, ed, ALs, ALd, M, DEN, WH, WA);
        post_kernel<<<NODES / 8, 256, 0, stream>>>(WX, WA, bias_l + l * DIM,
                                                   gamma + l * DIM, beta + l * DIM);
    }

    pool_init<<<(NGRAPH * DIM + 255) / 256, 256, 0, stream>>>(GS, CNT);
    pool_accum<<<(NODES * 32 + 255) / 256, 256, 0, stream>>>(WX, batch, GS, CNT);
    pool_final<<<(NGRAPH * DIM + 255) / 256, 256, 0, stream>>>(GS, CNT, (float*)d_out);
}